// LightweightConv_74904229643081
// MI455X (gfx1250) — compile-verified
//
#include <hip/hip_runtime.h>

#define B   8
#define C   1024
#define T   4096
#define H   16
#define KS  15
#define PAD 7
#define F   (H * KS)   // 240
#define BT  (B * T)    // 32768

#define KB   64        // K-block staged per TDM tile (rows of x)
#define NKB  (C / KB)  // 16 K-blocks
#define XP   80        // LDS row pitch in floats (64 data + 16 pad: bank-conflict free halves)

typedef __attribute__((ext_vector_type(2))) float v2f;
typedef __attribute__((ext_vector_type(8))) float v8f;

#if __has_builtin(__builtin_amdgcn_tensor_load_to_lds)
#define USE_TDM 1
typedef __attribute__((ext_vector_type(4))) unsigned int tdm_g0_t;
typedef __attribute__((ext_vector_type(8))) int          tdm_i8_t;
typedef __attribute__((ext_vector_type(4))) int          tdm_i4_t;

// DMA a 64(row) x 64(t) fp32 tile of x (row stride T) into LDS at lds_off,
// padding +16 DWORDs after every 64 DWORDs (row pitch XP=80 floats).
__device__ __forceinline__ void tdm_load_tile(const float* gsrc, unsigned lds_off) {
  const unsigned long long ga = (unsigned long long)(uintptr_t)gsrc;
  tdm_g0_t g0;
  g0[0] = 1u;                                        // count=1, user descriptor
  g0[1] = lds_off;                                   // LDS byte address
  g0[2] = (unsigned)(ga & 0xFFFFFFFFu);              // global_addr[31:0]
  g0[3] = (unsigned)((ga >> 32) & 0x01FFFFFFu)       // global_addr[56:32]
        | (2u << 30);                                // type=2 ("image")
  tdm_i8_t g1;
  g1[0] = (int)((2u << 16)                           // data_size = 4B
        | (1u << 20)                                 // pad_enable
        | (5u << 22)                                 // pad_interval: every 64 DWORDs
        | (15u << 25));                              // pad_amount: 16 DWORDs
  g1[1] = (int)(64u << 16);                          // tensor_dim0[15:0] = 64  (bits 63:48)
  g1[2] = (int)(64u << 16);                          // tensor_dim1[15:0] = 64  (bits 95:80)
  g1[3] = (int)(64u << 16);                          // tile_dim0 = 64          (bits 127:112)
  g1[4] = (int)64;                                   // tile_dim1 = 64          (bits 143:128)
  g1[5] = (int)T;                                    // tensor_dim0_stride = T  (bits 191:160)
  g1[6] = 0;
  g1[7] = 0;
  tdm_i4_t z4 = {0, 0, 0, 0};                        // groups 2/3 unused (2D tensor)
  tdm_i8_t z8 = {0, 0, 0, 0, 0, 0, 0, 0};
  __builtin_amdgcn_tensor_load_to_lds(g0, g1, z4, z4, z8, 0);
}
#endif

// ---------------------------------------------------------------------------
// Kernel 1: proj[f][bt] = sum_c W[f][c] * x[b][c][t]   (M=240, N=32768, K=1024)
// Workgroup = 15 waves = full M; each wave owns one 16-row f-tile and a
// 16x64 D tile (4 accumulators). x tile (64x64) staged in LDS once per
// K-block (TDM double-buffered), so x is read from HBM exactly once.
// ---------------------------------------------------------------------------
__global__ __launch_bounds__(480) void k_proj_wmma(const float* __restrict__ x,
                                                   const float* __restrict__ W,
                                                   float* __restrict__ proj) {
  __shared__ float xs[2][KB * XP];                   // 2 x 20 KB

  const int tid   = threadIdx.x;
  const int lane  = tid & 31;
  const int wave  = tid >> 5;                        // 0..14 -> f-tile
  const int strip = blockIdx.x;                      // 0..511 (64-wide bt strips)
  const int b     = strip >> 6;
  const int t0    = (strip & 63) * 64;

  const int half = lane >> 4;                        // 0: lanes 0-15, 1: lanes 16-31
  const int l15  = lane & 15;
  const int koff = half * 2;                         // upper half-wave holds K=+2,+3

  const int f0 = wave * 16;
  const float* xb = x + b * (C * T) + t0;            // tile origin column
  const float* wr = W + (f0 + l15) * C;              // A row for this lane

  v8f acc0 = {}, acc1 = {}, acc2 = {}, acc3 = {};

#if USE_TDM
  if (wave == 0) tdm_load_tile(xb, (unsigned)(uintptr_t)&xs[0][0]);
#endif

  for (int kb = 0; kb < NKB; ++kb) {
    const int buf = kb & 1;
#if USE_TDM
    if (wave == 0) __builtin_amdgcn_s_wait_tensorcnt(0);   // current tile landed
    __syncthreads();                                       // publish buffer
    if (wave == 0 && kb + 1 < NKB)
      tdm_load_tile(xb + (kb + 1) * KB * T,
                    (unsigned)(uintptr_t)&xs[buf ^ 1][0]); // prefetch next tile
#else
    for (int i = tid; i < KB * 64; i += 480) {
      const int r = i >> 6, cc = i & 63;
      xs[buf][r * XP + cc] = xb[(kb * KB + r) * T + cc];
    }
    __syncthreads();
#endif

    const float* wk   = wr + kb * KB;                // A: W row, current K-block
    const float* lbuf = &xs[buf][0];

    for (int kk = 0; kk < KB; kk += 4) {
      v2f a;
      a.x = wk[kk + koff];
      a.y = wk[kk + koff + 1];

      const float* lp = lbuf + (kk + koff) * XP + l15;
      v2f b0, b1, b2, b3;
      b0.x = lp[0];  b0.y = lp[XP];
      b1.x = lp[16]; b1.y = lp[XP + 16];
      b2.x = lp[32]; b2.y = lp[XP + 32];
      b3.x = lp[48]; b3.y = lp[XP + 48];

      acc0 = __builtin_amdgcn_wmma_f32_16x16x4_f32(false, a, false, b0, (short)0, acc0, false, false);
      acc1 = __builtin_amdgcn_wmma_f32_16x16x4_f32(false, a, false, b1, (short)0, acc1, false, false);
      acc2 = __builtin_amdgcn_wmma_f32_16x16x4_f32(false, a, false, b2, (short)0, acc2, false, false);
      acc3 = __builtin_amdgcn_wmma_f32_16x16x4_f32(false, a, false, b3, (short)0, acc3, false, false);
    }
    __syncthreads();                                 // done reading buf before overwrite
  }

  // D layout: VGPR v holds (row f0 + v + 8*half, col = l15) -> coalesced along bt
  const int btbase = strip * 64 + l15;
#pragma unroll
  for (int v = 0; v < 8; ++v) {
    const int f = f0 + v + half * 8;
    float* p = proj + f * BT + btbase;
    p[0]  = acc0[v];
    p[16] = acc1[v];
    p[32] = acc2[v];
    p[48] = acc3[v];
  }
}

// ---------------------------------------------------------------------------
// Kernel 2: in-place softmax over the 15 taps: proj[(h*15+k)][bt], k=0..14.
// ---------------------------------------------------------------------------
__global__ __launch_bounds__(256) void k_softmax(float* __restrict__ proj) {
  const int tid = blockIdx.x * 256 + threadIdx.x;    // 0 .. H*BT-1
  const int h   = tid >> 15;                         // / BT
  const int bt  = tid & (BT - 1);
  float* p = proj + (h * KS) * BT + bt;

  float v[KS];
  float m = -1e30f;
#pragma unroll
  for (int k = 0; k < KS; ++k) { v[k] = p[k * BT]; m = fmaxf(m, v[k]); }
  float s = 0.0f;
#pragma unroll
  for (int k = 0; k < KS; ++k) { v[k] = __expf(v[k] - m); s += v[k]; }
  const float inv = 1.0f / s;
#pragma unroll
  for (int k = 0; k < KS; ++k) { p[k * BT] = v[k] * inv; }
}

// ---------------------------------------------------------------------------
// Kernel 3: depthwise dynamic conv, LDS-tiled along t.
// ---------------------------------------------------------------------------
#define TB 256
__global__ __launch_bounds__(TB) void k_conv(const float* __restrict__ x,
                                             const float* __restrict__ w,  // softmaxed proj [F][BT]
                                             float* __restrict__ out) {
  __shared__ float xl[TB + KS - 1];
  const int tid = threadIdx.x;
  const int t0  = blockIdx.x * TB;
  const int c   = blockIdx.y;
  const int b   = blockIdx.z;
  const int h   = c & (H - 1);

  const float* xc = x + (b * C + c) * T;
  for (int i = tid; i < TB + KS - 1; i += TB) {
    const int tg = t0 + i - PAD;
    xl[i] = (tg >= 0 && tg < T) ? xc[tg] : 0.0f;
  }
  __syncthreads();

  const int t = t0 + tid;
  const float* wp = w + (h * KS) * BT + b * T + t;
  float acc = 0.0f;
#pragma unroll
  for (int k = 0; k < KS; ++k) {
    acc = fmaf(wp[k * BT], xl[tid + k], acc);
  }
  out[(b * C + c) * T + t] = acc;
}

// ---------------------------------------------------------------------------
extern "C" void kernel_launch(void* const* d_in, const int* in_sizes, int n_in,
                              void* d_out, int out_size, void* d_ws, size_t ws_size,
                              hipStream_t stream) {
  const float* x = (const float*)d_in[0];   // (B, C, T) fp32
  const float* W = (const float*)d_in[1];   // (H*K, C) fp32
  float* out  = (float*)d_out;              // (B, C, T) fp32
  float* proj = (float*)d_ws;               // [F][BT] = 240*32768 floats (31.5 MB)

  k_proj_wmma<<<dim3(BT / 64), 480, 0, stream>>>(x, W, proj);   // 512 WGs x 15 waves

  k_softmax<<<(H * BT) / 256, 256, 0, stream>>>(proj);

  k_conv<<<dim3(T / TB, C, B), TB, 0, stream>>>(x, proj, out);
}